// Conv1dSubsampling_39857296507476
// MI455X (gfx1250) — compile-verified
//
#include <hip/hip_runtime.h>
#include <hip/hip_bf16.h>

// ---------------------------------------------------------------------------
// Conv1dSubsampling fused kernel for gfx1250 (MI455X).
//   pointwise conv (K=80 GEMM) via V_WMMA_F32_16X16X4_F32, fused with the
//   depthwise k=8/stride=4 conv through LDS (avoids the 134MB intermediate).
//   Memory-bound: ~55MB mandatory traffic -> ~2.4us floor at 23.3 TB/s.
// ---------------------------------------------------------------------------

#define IN_CH   80
#define OUT_CH  512
#define T_IN    4096
#define SF      4
#define KW      8            // 2*SF
#define T_OUT   1023         // (4096 - 8)/4 + 1
#define BATCH   16
#define TW      80           // y-time window per wave (5 WMMA N-tiles)
#define TWP     81           // padded LDS stride (dwords) to avoid bank conflicts

#define W1Q_OFF 0
#define W2Q_OFF (OUT_CH * IN_CH)          // 40960 floats
#define Y_OUT_ELEMS ((size_t)BATCH * OUT_CH * T_OUT)  // 8,380,416

typedef __attribute__((ext_vector_type(2))) float v2f;
typedef __attribute__((ext_vector_type(8))) float v8f;

// ---------------------------------------------------------------------------
// Kernel 1: per-output-channel symmetric fake-quant of w1 (512x80) and
// w2 (512x8) into d_ws; also x_len_out = floor(x_len/4).
// One wave32 per channel; 64 blocks x 8 waves = 512 channels.
// ---------------------------------------------------------------------------
__global__ __launch_bounds__(256)
void quant_kernel(const float* __restrict__ w1,
                  const float* __restrict__ w2,
                  const int*   __restrict__ x_len,
                  float* __restrict__ ws,
                  float* __restrict__ out)
{
    const int lane = threadIdx.x & 31;
    const int wave = threadIdx.x >> 5;
    const int ch   = blockIdx.x * 8 + wave;   // 0..511

    // ---- w1 row: 80 elements, 3 strided per lane ----
    float v[3];
    float m = 0.f;
#pragma unroll
    for (int i = 0; i < 3; ++i) {
        int c = lane + 32 * i;
        v[i] = (c < IN_CH) ? w1[ch * IN_CH + c] : 0.f;
        m = fmaxf(m, fabsf(v[i]));
    }
#pragma unroll
    for (int off = 16; off >= 1; off >>= 1)
        m = fmaxf(m, __shfl_xor(m, off, 32));
    float s = fmaxf(m * (1.0f / 127.0f), 1e-5f);
#pragma unroll
    for (int i = 0; i < 3; ++i) {
        int c = lane + 32 * i;
        if (c < IN_CH)
            ws[W1Q_OFF + ch * IN_CH + c] = rintf(v[i] / s) * s;  // RNE == jnp.round
    }

    // ---- w2 row: 8 elements ----
    float w2v = (lane < KW) ? w2[ch * KW + lane] : 0.f;
    float m2 = fabsf(w2v);
#pragma unroll
    for (int off = 16; off >= 1; off >>= 1)
        m2 = fmaxf(m2, __shfl_xor(m2, off, 32));
    float s2 = fmaxf(m2 * (1.0f / 127.0f), 1e-5f);
    if (lane < KW)
        ws[W2Q_OFF + ch * KW + lane] = rintf(w2v / s2) * s2;

    // ---- x_len_out (second tuple output) ----
    if (blockIdx.x == 0 && threadIdx.x < BATCH)
        out[Y_OUT_ELEMS + threadIdx.x] = floorf((float)x_len[threadIdx.x] * 0.25f);
}

// ---------------------------------------------------------------------------
// Kernel 2: fused pointwise-GEMM (WMMA f32 16x16x4) + depthwise conv.
// Grid: (8 tprime-supertiles, 32 o-tiles, 16 batches), 256 threads = 8 waves.
// Each wave owns one 16(out-ch) x 16(t') output tile:
//   - accumulates y over an 80-wide time window: 20 K-steps x 5 N-tiles
//     = 100 v_wmma_f32_16x16x4_f32 per wave, accs held in 40 VGPRs
//   - OOB time handling via address clamp + multiply by a 0/1 float mask, so every
//     load issues unconditionally (no exec-mask branches in the hot loop)
//   - spills y-tile to LDS (stride TWP=81 dwords), then the 8-tap/stride-4
//     depthwise reduction reads LDS and stores to global.
// ---------------------------------------------------------------------------
__global__ __launch_bounds__(256)
void conv_fused_kernel(const float* __restrict__ x,
                       const float* __restrict__ ws,
                       float* __restrict__ out)
{
    __shared__ float lds[8 * 16 * TWP];   // 8 waves * 16ch * 81 dwords = 41472 B

    const int lane    = threadIdx.x & 31;
    const int wave    = threadIdx.x >> 5;
    const int tp_tile = blockIdx.x * 8 + wave;   // 0..63  (64*16 = 1024 >= 1023 t')
    const int o0      = blockIdx.y * 16;         // out-channel tile base
    const int b       = blockIdx.z;

    const int tp0 = tp_tile * 16;   // output t' base for this wave
    const int ty0 = tp0 * SF;       // y-time base (window [ty0, ty0+79])

    const int tloc  = lane & 15;          // N index (time) / M index (out-ch)
    const int khalf = (lane >> 4) << 1;   // 0 for lanes 0-15, 2 for lanes 16-31

    const float* w1q = ws + W1Q_OFF;
    const float* w2q = ws + W2Q_OFF;

    // Per-N-tile time index: clamped address (always safe to load) + zero mask.
    // K-invariant, so hoisted out of the K-loop entirely.
    int   tcl[5];
    float msk[5];
#pragma unroll
    for (int j = 0; j < 5; ++j) {
        int t  = ty0 + tloc + 16 * j;
        bool ok = (t < T_IN);
        tcl[j] = ok ? t : (T_IN - 1);
        msk[j] = ok ? 1.0f : 0.0f;
    }

    v8f acc[5] = {};   // 5 N-tiles of 16x16 f32 accumulators

    const float* xb   = x + (size_t)b * IN_CH * T_IN;
    const int    arow = (o0 + tloc) * IN_CH;

    // K loop over input channels, 4 at a time (exactly 20 steps for K=80).
    // Unroll x4 so the scheduler can batch loads ahead of the WMMAs.
#pragma unroll 4
    for (int c0 = 0; c0 < IN_CH; c0 += 4) {
        // A tile: w1q[o0+M, c0..c0+3] per ISA 16x4 f32 layout (b64 load)
        v2f a;
        a.x = w1q[arow + c0 + khalf];
        a.y = w1q[arow + c0 + khalf + 1];

        const float* xc0 = xb + (size_t)(c0 + khalf) * T_IN;
        const float* xc1 = xc0 + T_IN;

#pragma unroll
        for (int j = 0; j < 5; ++j) {
            // Unconditional loads (clamped address), zeroed by multiplying with the 0/1 mask.
            float b0 = xc0[tcl[j]];
            float b1 = xc1[tcl[j]];
            v2f bm;
            bm.x = b0 * msk[j];
            bm.y = b1 * msk[j];
            acc[j] = __builtin_amdgcn_wmma_f32_16x16x4_f32(
                /*neg_a=*/false, a, /*neg_b=*/false, bm,
                /*c_mod=*/(short)0, acc[j],
                /*reuse_a=*/false, /*reuse_b=*/false);
        }
    }

    // Spill y-tile to LDS: [ch 0..15][t_local 0..79], stride TWP.
    float* myl = lds + wave * 16 * TWP;
    const int chbase = (lane >> 4) << 3;  // C/D layout: upper lane-half = M+8
#pragma unroll
    for (int j = 0; j < 5; ++j) {
#pragma unroll
        for (int r = 0; r < 8; ++r) {
            myl[(chbase + r) * TWP + tloc + 16 * j] = acc[j][r];
        }
    }
    __syncthreads();   // each wave reads only its own region; barrier for DS order

    // Depthwise conv: 256 outputs per wave -> 8 per lane.
    // flat = q*32 + lane: tp = flat&15 (contiguous t' across a half-wave for
    // coalesced stores), ch = flat>>4.
#pragma unroll
    for (int q = 0; q < 8; ++q) {
        int flat = q * 32 + lane;
        int tp   = flat & 15;
        int ch   = flat >> 4;
        const float* wrow = w2q + (o0 + ch) * KW;
        const float* yrow = myl + ch * TWP + tp * SF;
        float sum = 0.f;
#pragma unroll
        for (int k = 0; k < KW; ++k)
            sum = fmaf(wrow[k], yrow[k], sum);
        int tprime = tp0 + tp;
        if (tprime < T_OUT)
            out[((size_t)b * OUT_CH + (o0 + ch)) * T_OUT + tprime] = sum;
    }
}

// ---------------------------------------------------------------------------
extern "C" void kernel_launch(void* const* d_in, const int* in_sizes, int n_in,
                              void* d_out, int out_size, void* d_ws, size_t ws_size,
                              hipStream_t stream) {
    const float* x     = (const float*)d_in[0];   // [16, 80, 4096] f32
    const int*   x_len = (const int*)  d_in[1];   // [16] i32
    const float* w1    = (const float*)d_in[2];   // [512, 80, 1] f32
    const float* w2    = (const float*)d_in[3];   // [512, 1, 8] f32
    float* out = (float*)d_out;                   // [16*512*1023] + [16] f32
    float* ws  = (float*)d_ws;                    // w1q (40960) + w2q (4096) floats

    quant_kernel<<<dim3(64), dim3(256), 0, stream>>>(w1, w2, x_len, ws, out);
    conv_fused_kernel<<<dim3(8, 32, 16), dim3(256), 0, stream>>>(x, ws, out);
}